// qlayer_24653112279148
// MI455X (gfx1250) — compile-verified
//
#include <hip/hip_runtime.h>
#include <math.h>

// ---------------------------------------------------------------------------
// Problem constants (reference: NDIM=256, BATCH=65536, NROT=32640)
// ---------------------------------------------------------------------------
#define NDIM   256
#define NROT   32640
#define M_TILE 32            // batch rows per workgroup in the fused kernel
#define S2     (NDIM + 1)    // padded LDS row stride (257) -> conflict-free col reads
#define C_PER_BLK 32         // columns per block in the rotation-builder
#define S1     (C_PER_BLK + 1)

typedef __attribute__((ext_vector_type(2))) float v2f;
typedef __attribute__((ext_vector_type(8))) float v8f;

// ---------------------------------------------------------------------------
// Kernel 1: build R1^T and R2^T from the Givens chains.
// Columns of M are independent under row-rotations, so: 1 thread = 1 column.
// The combinations(range(n),2) order keeps row i hot across the inner j sweep,
// so row i is register-carried: the sequential dependence collapses to a
// 2-FMA register chain per step; row-j LDS traffic pipelines around it.
// Output layout: RT[k*256 + n] = R[n][k]  (B-matrix friendly for out = z@R^T).
// ---------------------------------------------------------------------------
__global__ __launch_bounds__(C_PER_BLK) void build_rot_kernel(
    const float* __restrict__ rots1,
    const float* __restrict__ rots2,
    float* __restrict__ rt_out)           // [2][256*256]
{
    __shared__ float M[NDIM * S1];        // 256 x 33 floats = 33 KB

    const int blk    = blockIdx.x;        // 0..15
    const int mat    = blk >> 3;          // 0: R1, 1: R2
    const int colgrp = blk & 7;           // which 32-column slice
    const float* __restrict__ th = mat ? rots2 : rots1;
    float* __restrict__ RT = rt_out + (size_t)mat * NDIM * NDIM;

    const int t   = threadIdx.x;          // 0..31 local column
    const int col = colgrp * C_PER_BLK + t;

    // identity columns
    for (int r = 0; r < NDIM; ++r)
        M[r * S1 + t] = (r == col) ? 1.0f : 0.0f;
    __syncthreads();

    int k = 0;
    for (int i = 0; i < NDIM - 1; ++i) {
        float ri = M[i * S1 + t];                   // register-carried row i
        for (int j = i + 1; j < NDIM; ++j, ++k) {
            const float thv = th[k];                // scalar-uniform load
            const float c = cosf(thv);
            const float s = sinf(thv);
            const float rj = M[j * S1 + t];
            M[j * S1 + t] = s * ri + c * rj;        // row j touched once
            ri = c * ri - s * rj;                   // short FMA chain
        }
        M[i * S1 + t] = ri;
    }
    __syncthreads();

    // Thread t holds R[.][col]  ==  RT[col][.].  Emit coalesced along n via
    // an LDS transpose: for matrix-row q of this slice, 32 lanes write 32
    // consecutive n. Bank pattern (n*33+q): 33 odd -> conflict-free.
    for (int q = 0; q < C_PER_BLK; ++q) {
        const int rr = colgrp * C_PER_BLK + q;      // RT row (= k index)
        for (int nb = 0; nb < NDIM / C_PER_BLK; ++nb) {
            const int n = nb * C_PER_BLK + t;
            RT[(size_t)rr * NDIM + n] = M[n * S1 + q];
        }
    }
}

// ---------------------------------------------------------------------------
// Kernel 2: fused  normalize -> GEMM1 -> phase -> GEMM2(x2) -> |z|^2
// 32 batch rows per WG, 8 waves; each wave owns a 2x2 grid of 16x16 output
// tiles (2 M-tiles x 2 N-tiles) so each B fragment feeds 2 wmmas (GEMM1) /
// 4 wmmas (GEMM2, re+im share B), halving VMEM+DS issue per wmma.
// fp32 WMMA 16x16x4; K-loop of 64 steps covers K=256.
//   A frag (16x4, 2 VGPR): lane&15 = M, half-wave selects K pair (ISA layout)
//   B frag (4x16, 2 VGPR): lane&15 = N, half-wave selects K pair
//   C/D   (16x16, 8 VGPR): VGPR v -> M = v + 8*(lane>>4), N = lane&15
// ---------------------------------------------------------------------------
__global__ __launch_bounds__(256) void qlayer_fused_kernel(
    const float* __restrict__ x,          // [BATCH, 256]
    const float* __restrict__ phases,     // [256]
    const float* __restrict__ rt1,        // R1^T, [256,256]
    const float* __restrict__ rt2,        // R2^T, [256,256]
    float2* __restrict__ out)             // [BATCH, 256] complex64
{
    __shared__ float zn[M_TILE * S2];     // normalized x tile   (~32.1 KB)
    __shared__ float wr[M_TILE * S2];     // Re(z) after phase
    __shared__ float wi[M_TILE * S2];     // Im(z) after phase   (total ~96.4 KB)

    const int tid  = threadIdx.x;
    const int lane = tid & 31;
    const int wave = tid >> 5;            // 0..7
    const int half = lane >> 4;           // 0/1 (K-pair select)
    const int l15  = lane & 15;           // M (A,C) or N (B,C)
    const size_t mb = (size_t)blockIdx.x * M_TILE;

    // ---- load x tile --------------------------------------------------
    for (int idx = tid; idx < M_TILE * NDIM; idx += 256) {
        const int r = idx >> 8, c = idx & 255;
        zn[r * S2 + c] = x[mb * NDIM + idx];
    }
    __syncthreads();

    // ---- row L2 norms (stash rnorm in wr[0..31], consumed before reuse)
    if (tid < M_TILE) {
        float s = 0.0f;
        for (int c = 0; c < NDIM; ++c) {
            const float v = zn[tid * S2 + c];
            s += v * v;
        }
        wr[tid] = rsqrtf(s);
    }
    __syncthreads();
    for (int idx = tid; idx < M_TILE * NDIM; idx += 256) {
        const int r = idx >> 8, c = idx & 255;
        zn[r * S2 + c] *= wr[r];
    }
    __syncthreads();

    const int nb0 = wave * 32;            // first N-tile base
    const int nb1 = nb0 + 16;             // second N-tile base

    // ---- GEMM1 (zn @ R1^T): 2x2 tiles per wave ------------------------
    {
        v8f acc00 = {}, acc01 = {}, acc10 = {}, acc11 = {};
        for (int k = 0; k < NDIM; k += 4) {
            const int ko = k + 2 * half;
            v2f a0, a1, b0, b1;
            a0.x = zn[l15 * S2 + ko];
            a0.y = zn[l15 * S2 + ko + 1];
            a1.x = zn[(16 + l15) * S2 + ko];
            a1.y = zn[(16 + l15) * S2 + ko + 1];
            b0.x = rt1[(size_t)ko * NDIM + nb0 + l15];
            b0.y = rt1[(size_t)(ko + 1) * NDIM + nb0 + l15];
            b1.x = rt1[(size_t)ko * NDIM + nb1 + l15];
            b1.y = rt1[(size_t)(ko + 1) * NDIM + nb1 + l15];
            acc00 = __builtin_amdgcn_wmma_f32_16x16x4_f32(false, a0, false, b0, (short)0, acc00, false, false);
            acc01 = __builtin_amdgcn_wmma_f32_16x16x4_f32(false, a0, false, b1, (short)0, acc01, false, false);
            acc10 = __builtin_amdgcn_wmma_f32_16x16x4_f32(false, a1, false, b0, (short)0, acc10, false, false);
            acc11 = __builtin_amdgcn_wmma_f32_16x16x4_f32(false, a1, false, b1, (short)0, acc11, false, false);
        }
        // phase rotation on accumulators (per-lane N fixed per tile column)
        const float ph0 = phases[nb0 + l15];
        const float ph1 = phases[nb1 + l15];
        const float cp0 = __cosf(ph0), sp0 = __sinf(ph0);
        const float cp1 = __cosf(ph1), sp1 = __sinf(ph1);
#pragma unroll
        for (int v = 0; v < 8; ++v) {
            const int m0 = v + 8 * half;
            const int m1 = 16 + m0;
            wr[m0 * S2 + nb0 + l15] = acc00[v] * cp0;
            wi[m0 * S2 + nb0 + l15] = acc00[v] * sp0;
            wr[m0 * S2 + nb1 + l15] = acc01[v] * cp1;
            wi[m0 * S2 + nb1 + l15] = acc01[v] * sp1;
            wr[m1 * S2 + nb0 + l15] = acc10[v] * cp0;
            wi[m1 * S2 + nb0 + l15] = acc10[v] * sp0;
            wr[m1 * S2 + nb1 + l15] = acc11[v] * cp1;
            wi[m1 * S2 + nb1 + l15] = acc11[v] * sp1;
        }
    }
    __syncthreads();   // all of wr/wi written before any wave reads full K

    // ---- GEMM2 (re & im share B frags) + |z|^2 ------------------------
    {
        v8f ar00 = {}, ar01 = {}, ar10 = {}, ar11 = {};
        v8f ai00 = {}, ai01 = {}, ai10 = {}, ai11 = {};
        for (int k = 0; k < NDIM; k += 4) {
            const int ko = k + 2 * half;
            v2f b0, b1, r0, r1, i0, i1;
            b0.x = rt2[(size_t)ko * NDIM + nb0 + l15];
            b0.y = rt2[(size_t)(ko + 1) * NDIM + nb0 + l15];
            b1.x = rt2[(size_t)ko * NDIM + nb1 + l15];
            b1.y = rt2[(size_t)(ko + 1) * NDIM + nb1 + l15];
            r0.x = wr[l15 * S2 + ko];
            r0.y = wr[l15 * S2 + ko + 1];
            r1.x = wr[(16 + l15) * S2 + ko];
            r1.y = wr[(16 + l15) * S2 + ko + 1];
            i0.x = wi[l15 * S2 + ko];
            i0.y = wi[l15 * S2 + ko + 1];
            i1.x = wi[(16 + l15) * S2 + ko];
            i1.y = wi[(16 + l15) * S2 + ko + 1];
            ar00 = __builtin_amdgcn_wmma_f32_16x16x4_f32(false, r0, false, b0, (short)0, ar00, false, false);
            ar01 = __builtin_amdgcn_wmma_f32_16x16x4_f32(false, r0, false, b1, (short)0, ar01, false, false);
            ai00 = __builtin_amdgcn_wmma_f32_16x16x4_f32(false, i0, false, b0, (short)0, ai00, false, false);
            ai01 = __builtin_amdgcn_wmma_f32_16x16x4_f32(false, i0, false, b1, (short)0, ai01, false, false);
            ar10 = __builtin_amdgcn_wmma_f32_16x16x4_f32(false, r1, false, b0, (short)0, ar10, false, false);
            ar11 = __builtin_amdgcn_wmma_f32_16x16x4_f32(false, r1, false, b1, (short)0, ar11, false, false);
            ai10 = __builtin_amdgcn_wmma_f32_16x16x4_f32(false, i1, false, b0, (short)0, ai10, false, false);
            ai11 = __builtin_amdgcn_wmma_f32_16x16x4_f32(false, i1, false, b1, (short)0, ai11, false, false);
        }
#pragma unroll
        for (int v = 0; v < 8; ++v) {
            const int m0 = v + 8 * half;
            const int m1 = 16 + m0;
            out[(mb + m0) * NDIM + nb0 + l15] =
                make_float2(ar00[v] * ar00[v] + ai00[v] * ai00[v], 0.0f);
            out[(mb + m0) * NDIM + nb1 + l15] =
                make_float2(ar01[v] * ar01[v] + ai01[v] * ai01[v], 0.0f);
            out[(mb + m1) * NDIM + nb0 + l15] =
                make_float2(ar10[v] * ar10[v] + ai10[v] * ai10[v], 0.0f);
            out[(mb + m1) * NDIM + nb1 + l15] =
                make_float2(ar11[v] * ar11[v] + ai11[v] * ai11[v], 0.0f);
        }
    }
}

// ---------------------------------------------------------------------------
// Host launcher. Inputs: [0]=x, [1]=rots1, [2]=phases, [3]=rots2.
// Workspace: RT1 at ws[0..65535], RT2 at ws[65536..131071] (512 KB total).
// Same-stream ordering makes kernel2 see the completed rotation matrices.
// ---------------------------------------------------------------------------
extern "C" void kernel_launch(void* const* d_in, const int* in_sizes, int n_in,
                              void* d_out, int out_size, void* d_ws, size_t ws_size,
                              hipStream_t stream) {
    (void)n_in; (void)out_size; (void)ws_size;
    const float* x      = (const float*)d_in[0];
    const float* rots1  = (const float*)d_in[1];
    const float* phases = (const float*)d_in[2];
    const float* rots2  = (const float*)d_in[3];
    float* rt = (float*)d_ws;

    build_rot_kernel<<<16, C_PER_BLK, 0, stream>>>(rots1, rots2, rt);

    const int batch = in_sizes[0] / NDIM;   // 65536
    qlayer_fused_kernel<<<batch / M_TILE, 256, 0, stream>>>(
        x, phases, rt, rt + NDIM * NDIM, (float2*)d_out);
}